// GAT_12498354831961
// MI455X (gfx1250) — compile-verified
//
#include <hip/hip_runtime.h>
#include <hip/hip_bf16.h>

// GAT (2 layers) for MI455X / gfx1250.
// GEMM1 (100000x128 @ 128x256) on the matrix pipe via V_WMMA_F32_16X16X4_F32,
// 2x2 register tiling (4 accumulators/wave) for B-fragment reuse, and async
// Global->LDS (GLOBAL_LOAD_ASYNC_TO_LDS_B128 / ASYNCcnt) for the A tile.
// The LDS address handed to the async op is derived from As's own pointer
// (low 32 bits of the flat address == LDS offset) so the compiler sees As
// escape and cannot fold its reads to undef.
// Edge softmax: two atomic passes (ordered-uint atomicMax, then
// exp/sum/weighted aggregate of unnormalized numerators) + normalize pass.

#define NN      100000
#define EE      1600000
#define ET      (EE + NN)      // 1700000 edges incl. self loops
#define INC     128
#define HID     64
#define HEADS   4
#define F1      (HEADS * HID)  // 256
#define NEG_SL  0.2f

typedef float v2f __attribute__((ext_vector_type(2)));
typedef float v8f __attribute__((ext_vector_type(8)));

__device__ __forceinline__ unsigned fenc(float f) {
  unsigned u = __float_as_uint(f);
  return (u & 0x80000000u) ? ~u : (u | 0x80000000u);
}
__device__ __forceinline__ float fdec(unsigned e) {
  return (e & 0x80000000u) ? __uint_as_float(e & 0x7FFFFFFFu)
                           : __uint_as_float(~e);
}
__device__ __forceinline__ float lrelu(float v) { return v > 0.f ? v : NEG_SL * v; }

__device__ __forceinline__ void edge_sd(const int* __restrict__ ei, int e, int& s, int& d) {
  if (e < EE) { s = ei[e]; d = ei[EE + e]; }
  else        { s = e - EE; d = e - EE; }      // self loops appended
}

// ---------------------------------------------------------------- GEMM1 (WMMA)
// h1[N,256] = x[N,128] @ W1[128,256].  One block = one 32-row M tile.
// A tile (32x128 f32 = 16KB) brought in with async Global->LDS; each of the
// 8 waves owns a 32x32 output patch via a 2x2 grid of 16x16 accumulators.
__global__ __launch_bounds__(256) void gat_gemm1(const float* __restrict__ x,
                                                 const float* __restrict__ W,
                                                 float* __restrict__ h1) {
  __shared__ __align__(16) float As[32 * INC];   // 16 KB
  const int tid  = threadIdx.x;
  const int row0 = blockIdx.x * 32;              // 100000 % 32 == 0 -> no tail

  // Async copy x[row0..row0+31][0..127] -> LDS. 16KB / (256 thr * 16B) = 4 issues.
  // lds offset = low 32 bits of As's flat address (aperture holds the high
  // bits on gfx1250); ptrtoint also makes As escape so the "memory" clobber
  // is honored for it.
  {
    const unsigned lds_base = (unsigned)(size_t)(void*)As;
    const char* gsrc = (const char*)(x + (size_t)row0 * INC);
    #pragma unroll
    for (int i = 0; i < 4; ++i) {
      unsigned           ldsoff = lds_base + (unsigned)(tid * 16 + i * 4096);
      unsigned long long gaddr  = (unsigned long long)(gsrc + tid * 16 + i * 4096);
      asm volatile("global_load_async_to_lds_b128 %0, %1, off"
                   :: "v"(ldsoff), "v"(gaddr) : "memory");
    }
    asm volatile("s_wait_asynccnt 0x0" ::: "memory");
  }
  __syncthreads();

  const int wave = tid >> 5;
  const int lane = tid & 31;
  const int lh   = lane >> 4;                    // lane half (0/1)
  const int l16  = lane & 15;
  const int col0 = wave * 32;

  v8f a00 = {}, a01 = {}, a10 = {}, a11 = {};
  for (int kk = 0; kk < INC; kk += 4) {
    const int ka = kk + 2 * lh;                  // A: lanes0-15 K=kk,kk+1; lanes16-31 K=kk+2,kk+3
    v2f fa0, fa1, fb0, fb1;
    fa0.x = As[l16 * INC + ka];
    fa0.y = As[l16 * INC + ka + 1];
    fa1.x = As[(16 + l16) * INC + ka];
    fa1.y = As[(16 + l16) * INC + ka + 1];
    fb0.x = W[(size_t)ka * F1 + col0 + l16];
    fb0.y = W[(size_t)(ka + 1) * F1 + col0 + l16];
    fb1.x = W[(size_t)ka * F1 + col0 + 16 + l16];
    fb1.y = W[(size_t)(ka + 1) * F1 + col0 + 16 + l16];
    a00 = __builtin_amdgcn_wmma_f32_16x16x4_f32(false, fa0, false, fb0, (short)0, a00, false, false);
    a01 = __builtin_amdgcn_wmma_f32_16x16x4_f32(false, fa0, false, fb1, (short)0, a01, false, false);
    a10 = __builtin_amdgcn_wmma_f32_16x16x4_f32(false, fa1, false, fb0, (short)0, a10, false, false);
    a11 = __builtin_amdgcn_wmma_f32_16x16x4_f32(false, fa1, false, fb1, (short)0, a11, false, false);
  }
  // C/D layout: VGPR r, lanes0-15 -> (M=r, N=lane); lanes16-31 -> (M=r+8, N=lane-16)
  #pragma unroll
  for (int r = 0; r < 8; ++r) {
    const int m0 = row0 + r + 8 * lh;
    const int m1 = m0 + 16;
    h1[(size_t)m0 * F1 + col0 + l16]      = a00[r];
    h1[(size_t)m0 * F1 + col0 + 16 + l16] = a01[r];
    h1[(size_t)m1 * F1 + col0 + l16]      = a10[r];
    h1[(size_t)m1 * F1 + col0 + 16 + l16] = a11[r];
  }
}

// ------------------------------------------------------- per-node alpha (L1)
__global__ __launch_bounds__(256) void gat_alpha1(const float* __restrict__ h1,
                                                  const float* __restrict__ a_src,
                                                  const float* __restrict__ a_dst,
                                                  float* __restrict__ as,
                                                  float* __restrict__ ad) {
  int idx = blockIdx.x * 256 + threadIdx.x;
  if (idx >= NN * HEADS) return;
  int n = idx >> 2, h = idx & 3;
  const float* row = h1 + (size_t)n * F1 + h * HID;
  const float* vs = a_src + h * HID;
  const float* vd = a_dst + h * HID;
  float ss = 0.f, dd = 0.f;
  #pragma unroll 8
  for (int i = 0; i < HID; ++i) { float v = row[i]; ss += v * vs[i]; dd += v * vd[i]; }
  as[idx] = ss;
  ad[idx] = dd;
}

// ------------------------------------------------------- edge pass A (L1 max)
__global__ __launch_bounds__(256) void gat_edge_max1(const int* __restrict__ ei,
                                                     const float* __restrict__ as,
                                                     const float* __restrict__ ad,
                                                     unsigned* __restrict__ m1) {
  int idx = blockIdx.x * 256 + threadIdx.x;
  if (idx >= ET * HEADS) return;
  int e = idx >> 2, h = idx & 3;
  int s, d; edge_sd(ei, e, s, d);
  float v = lrelu(as[s * 4 + h] + ad[d * 4 + h]);
  atomicMax(&m1[d * 4 + h], fenc(v));
}

// ------------------------------------- edge pass B (L1 exp + sum + aggregate)
// One wave per edge: coalesced 256-channel gather of h1[src] and coalesced
// atomic scatter-add into out1[dst] (unnormalized numerator).
__global__ __launch_bounds__(256) void gat_edge_agg1(const int* __restrict__ ei,
                                                     const float* __restrict__ as,
                                                     const float* __restrict__ ad,
                                                     const unsigned* __restrict__ m1,
                                                     const float* __restrict__ h1,
                                                     float* __restrict__ s1,
                                                     float* __restrict__ out1) {
  int gid  = blockIdx.x * 256 + threadIdx.x;
  int e    = gid >> 5;
  int lane = gid & 31;
  if (e >= ET) return;
  int s, d; edge_sd(ei, e, s, d);
  const float* hrow = h1 + (size_t)s * F1;
  float*       orow = out1 + (size_t)d * F1;
  // Overlap the gather with the expf dependency chain (global_prefetch_b8).
  __builtin_prefetch(hrow + lane, 0, 3);
  __builtin_prefetch(hrow + 128 + lane, 0, 3);
  float ex[HEADS];
  #pragma unroll
  for (int h = 0; h < HEADS; ++h) {
    float v = lrelu(as[s * 4 + h] + ad[d * 4 + h]);
    ex[h] = expf(v - fdec(m1[d * 4 + h]));
  }
  if (lane == 0) {
    #pragma unroll
    for (int h = 0; h < HEADS; ++h) atomicAdd(&s1[d * 4 + h], ex[h]);
  }
  #pragma unroll
  for (int j = 0; j < 8; ++j) {                  // channel c = lane + 32j; head = j>>1
    int c = lane + 32 * j;
    atomicAdd(&orow[c], ex[j >> 1] * hrow[c]);
  }
}

// ------------------------------------------ normalize + bias + ELU (L1 -> act)
__global__ __launch_bounds__(256) void gat_fin1(float* __restrict__ out1,
                                                const float* __restrict__ s1,
                                                const float* __restrict__ b1) {
  int idx = blockIdx.x * 256 + threadIdx.x;
  if (idx >= NN * F1) return;
  int n = idx >> 8;
  int c = idx & 255;
  float v = out1[idx] / s1[n * 4 + (c >> 6)] + b1[c];
  out1[idx] = v > 0.f ? v : expm1f(v);           // ELU
}

// ---------------------------------- layer 2: h2 = act @ W2 (256->2) + alphas
__global__ __launch_bounds__(256) void gat_node2(const float* __restrict__ hact,
                                                 const float* __restrict__ W2,
                                                 const float* __restrict__ av_s,
                                                 const float* __restrict__ av_d,
                                                 float* __restrict__ h2,
                                                 float* __restrict__ as2,
                                                 float* __restrict__ ad2) {
  int gid  = blockIdx.x * 256 + threadIdx.x;
  int n    = gid >> 5;
  int lane = gid & 31;
  if (n >= NN) return;
  const float* row = hact + (size_t)n * F1;
  float p0 = 0.f, p1 = 0.f;
  #pragma unroll
  for (int j = 0; j < 8; ++j) {
    int c = lane + 32 * j;
    float v = row[c];
    p0 += v * W2[c * 2 + 0];
    p1 += v * W2[c * 2 + 1];
  }
  #pragma unroll
  for (int off = 16; off > 0; off >>= 1) {
    p0 += __shfl_xor(p0, off, 32);
    p1 += __shfl_xor(p1, off, 32);
  }
  if (lane == 0) {
    h2[n * 2 + 0] = p0;
    h2[n * 2 + 1] = p1;
    as2[n] = p0 * av_s[0] + p1 * av_s[1];
    ad2[n] = p0 * av_d[0] + p1 * av_d[1];
  }
}

__global__ __launch_bounds__(256) void gat_edge_max2(const int* __restrict__ ei,
                                                     const float* __restrict__ as,
                                                     const float* __restrict__ ad,
                                                     unsigned* __restrict__ m2) {
  int e = blockIdx.x * 256 + threadIdx.x;
  if (e >= ET) return;
  int s, d; edge_sd(ei, e, s, d);
  atomicMax(&m2[d], fenc(lrelu(as[s] + ad[d])));
}

__global__ __launch_bounds__(256) void gat_edge_agg2(const int* __restrict__ ei,
                                                     const float* __restrict__ as,
                                                     const float* __restrict__ ad,
                                                     const unsigned* __restrict__ m2,
                                                     const float* __restrict__ h2,
                                                     float* __restrict__ s2,
                                                     float* __restrict__ out2) {
  int e = blockIdx.x * 256 + threadIdx.x;
  if (e >= ET) return;
  int s, d; edge_sd(ei, e, s, d);
  float ex = expf(lrelu(as[s] + ad[d]) - fdec(m2[d]));
  atomicAdd(&s2[d], ex);
  atomicAdd(&out2[d * 2 + 0], ex * h2[s * 2 + 0]);
  atomicAdd(&out2[d * 2 + 1], ex * h2[s * 2 + 1]);
}

__global__ __launch_bounds__(256) void gat_fin2(const float* __restrict__ out2,
                                                const float* __restrict__ s2,
                                                const float* __restrict__ b2,
                                                float* __restrict__ out) {
  int idx = blockIdx.x * 256 + threadIdx.x;
  if (idx >= NN * 2) return;
  int n = idx >> 1, c = idx & 1;
  out[idx] = out2[idx] / s2[n] + b2[c];          // mean over 1 head == identity
}

// ------------------------------------------------------------------- launcher
extern "C" void kernel_launch(void* const* d_in, const int* in_sizes, int n_in,
                              void* d_out, int out_size, void* d_ws, size_t ws_size,
                              hipStream_t stream) {
  (void)in_sizes; (void)n_in; (void)out_size; (void)ws_size;
  const float* x      = (const float*)d_in[0];
  const int*   ei     = (const int*)d_in[1];     // [2,E]: row0=src, row1=dst
  const float* W1     = (const float*)d_in[2];
  const float* a_src1 = (const float*)d_in[3];
  const float* a_dst1 = (const float*)d_in[4];
  const float* b1     = (const float*)d_in[5];
  const float* W2     = (const float*)d_in[6];
  const float* a_src2 = (const float*)d_in[7];
  const float* a_dst2 = (const float*)d_in[8];
  const float* b2     = (const float*)d_in[9];
  float* out = (float*)d_out;

  float* ws = (float*)d_ws;
  size_t o = 0;
  float*    h1   = ws + o;               o += (size_t)NN * F1;
  float*    out1 = ws + o;               o += (size_t)NN * F1;
  float*    as1  = ws + o;               o += (size_t)NN * HEADS;
  float*    ad1  = ws + o;               o += (size_t)NN * HEADS;
  unsigned* m1   = (unsigned*)(ws + o);  o += (size_t)NN * HEADS;
  float*    s1   = ws + o;               o += (size_t)NN * HEADS;
  float*    h2   = ws + o;               o += (size_t)NN * 2;
  float*    as2  = ws + o;               o += (size_t)NN;
  float*    ad2  = ws + o;               o += (size_t)NN;
  unsigned* m2   = (unsigned*)(ws + o);  o += (size_t)NN;
  float*    s2   = ws + o;               o += (size_t)NN;
  float*    out2 = ws + o;               o += (size_t)NN * 2;

  // Accumulators must be re-zeroed every call (deterministic replay).
  hipMemsetAsync(out1, 0, (size_t)NN * F1 * sizeof(float), stream);
  hipMemsetAsync(m1,   0, (size_t)NN * HEADS * sizeof(unsigned), stream);
  hipMemsetAsync(s1,   0, (size_t)NN * HEADS * sizeof(float), stream);
  hipMemsetAsync(m2,   0, (size_t)NN * sizeof(unsigned), stream);
  hipMemsetAsync(s2,   0, (size_t)NN * sizeof(float), stream);
  hipMemsetAsync(out2, 0, (size_t)NN * 2 * sizeof(float), stream);

  gat_gemm1    <<<NN / 32,                   256, 0, stream>>>(x, W1, h1);
  gat_alpha1   <<<(NN * HEADS + 255) / 256,  256, 0, stream>>>(h1, a_src1, a_dst1, as1, ad1);
  gat_edge_max1<<<(ET * HEADS + 255) / 256,  256, 0, stream>>>(ei, as1, ad1, m1);
  gat_edge_agg1<<<(ET * 32 + 255) / 256,     256, 0, stream>>>(ei, as1, ad1, m1, h1, s1, out1);
  gat_fin1     <<<(NN * F1 + 255) / 256,     256, 0, stream>>>(out1, s1, b1);
  gat_node2    <<<(NN * 32 + 255) / 256,     256, 0, stream>>>(out1, W2, a_src2, a_dst2, h2, as2, ad2);
  gat_edge_max2<<<(ET + 255) / 256,          256, 0, stream>>>(ei, as2, ad2, m2);
  gat_edge_agg2<<<(ET + 255) / 256,          256, 0, stream>>>(ei, as2, ad2, m2, h2, s2, out2);
  gat_fin2     <<<(NN * 2 + 255) / 256,      256, 0, stream>>>(out2, s2, b2, out);
}